// RankPool1d_87308095193625
// MI455X (gfx1250) — compile-verified
//
#include <hip/hip_runtime.h>
#include <stdint.h>

#define BLOCK 256
#define KS    16
#define PADW  8
#define TILE  (BLOCK + KS)   // 272 floats = 1088 bytes per block tile
#define RNK   7              // 0-indexed rank (8th smallest)

// Compare-and-swap on values only: 2 VALU (v_min_num_f32 / v_max_num_f32)
__device__ __forceinline__ void cas(float& x, float& y) {
    float lo = fminf(x, y);
    float hi = fmaxf(x, y);
    x = lo; y = hi;
}

// Batcher odd-even mergesort for 8 elements: 19 comparators (optimal-known).
__device__ __forceinline__ void sort8(float* a) {
    cas(a[0],a[1]); cas(a[2],a[3]); cas(a[4],a[5]); cas(a[6],a[7]);
    cas(a[0],a[2]); cas(a[1],a[3]); cas(a[4],a[6]); cas(a[5],a[7]);
    cas(a[1],a[2]); cas(a[5],a[6]);
    cas(a[0],a[4]); cas(a[1],a[5]); cas(a[2],a[6]); cas(a[3],a[7]);
    cas(a[2],a[4]); cas(a[3],a[5]);
    cas(a[1],a[2]); cas(a[3],a[4]); cas(a[5],a[6]);
}

__global__ void __launch_bounds__(BLOCK)
rankpool1d_kernel(const float* __restrict__ x, float* __restrict__ out,
                  int L, int out_len) {
    __shared__ __align__(16) float smem[TILE];

    const int  t  = threadIdx.x;
    const long o0 = (long)blockIdx.x * BLOCK;   // first output of this block
    const long g0 = o0 - PADW;                  // first input element of tile

    const bool interior = (g0 >= 0) && (g0 + TILE <= (long)L);

    if (interior) {
        // Async DMA the whole tile into LDS: 68 lanes x b128 = 1088 bytes.
        // 16B alignment holds: g0*4 = 1024*blockIdx - 32.
        if (t < TILE / 4) {
            uint32_t lds  = (uint32_t)(uintptr_t)smem + (uint32_t)t * 16u;
            uint32_t voff = (uint32_t)((g0 + (long)t * 4) * 4);
#if defined(__gfx1250__)
            asm volatile("global_load_async_to_lds_b128 %0, %1, %2"
                         :
                         : "v"(lds), "v"(voff), "s"((uint64_t)(uintptr_t)x)
                         : "memory");
#endif
        }
#if defined(__gfx1250__)
        asm volatile("s_wait_asynccnt 0" ::: "memory");
#endif
    } else {
        // Boundary blocks: guarded fill with zero padding.
        for (int j = t; j < TILE; j += BLOCK) {
            long gi = g0 + j;
            smem[j] = (gi >= 0 && gi < (long)L) ? x[gi] : 0.0f;
        }
    }
    __syncthreads();

    const long o = o0 + t;
    if (o >= (long)out_len) return;

    // Window for output o lives at smem[t .. t+15] (bank-conflict-free).
    float w[16];
    #pragma unroll
    for (int k = 0; k < 16; ++k) w[k] = smem[t + k];

    // Sort the two halves, then the 8th-smallest of the 16 is
    // max_i min(a[i], b[7-i])  (bitonic lower-half split identity).
    float a[8], c[8];
    #pragma unroll
    for (int k = 0; k < 8; ++k) { a[k] = w[k]; c[k] = w[k + 8]; }
    sort8(a);
    sort8(c);

    float v = fminf(a[0], c[7]);
    #pragma unroll
    for (int i = 1; i < 8; ++i) v = fmaxf(v, fminf(a[i], c[7 - i]));

    // Stable-argsort index: among equal values, earlier index wins.
    // sel = (RNK - cntLess)-th occurrence of v in the window.
    int cnt = 0;
    #pragma unroll
    for (int k = 0; k < 16; ++k) cnt += (w[k] < v) ? 1 : 0;

    int need = RNK - cnt;
    int sel  = 0;
    #pragma unroll
    for (int k = 0; k < 16; ++k) {
        bool eq = (w[k] == v);
        if (eq && need == 0) sel = k;
        need -= eq ? 1 : 0;
    }

    out[o]           = v;            // values
    out[out_len + o] = (float)sel;   // indices (exact small ints in f32)
}

extern "C" void kernel_launch(void* const* d_in, const int* in_sizes, int n_in,
                              void* d_out, int out_size, void* d_ws, size_t ws_size,
                              hipStream_t stream) {
    const float* x   = (const float*)d_in[0];
    float*       out = (float*)d_out;
    const int L       = in_sizes[0];
    const int out_len = (L + 2 * PADW - KS) / 1 + 1;   // = L + 1
    const int blocks  = (out_len + BLOCK - 1) / BLOCK;
    rankpool1d_kernel<<<blocks, BLOCK, 0, stream>>>(x, out, L, out_len);
}